// SegmentationOHEMLoss_17643725652478
// MI455X (gfx1250) — compile-verified
//
#include <hip/hip_runtime.h>
#include <stdint.h>

// ---------------- problem constants (from reference) ----------------
#define NEG_POS 3
#define B_   16
#define C_   4
#define H_   512
#define W_   512
#define HW   (H_*W_)          // 262144 pixels per (b,c) slice
#define NSLICE (B_*C_)        // 64 contiguous slices in NCHW layout
#define NBINS  2048           // float-bit histogram: bits[30:19], max 0x3F800000>>19 = 2032
#define NBLK   16             // blocks per slice for the data passes
#define CHUNK  (HW/NBLK)      // 16384 elements per block
#define TPB    256            // 8 wave32s per block
#define BPT    (NBINS/TPB)    // 8 bins per thread in the threshold kernel

typedef float v2f __attribute__((ext_vector_type(2)));
typedef float v8f __attribute__((ext_vector_type(8)));

// ---------------- workspace layout (~1.06 MB, zeroed each launch) ----------------
struct WS {
  unsigned histC[NSLICE][NBINS];      // per-slice key-bin counts
  float    histS[NSLICE][NBINS];      // per-slice per-bin smoothL1 sums
  unsigned numPos[NSLICE];            // int atomics -> deterministic
  float    posPart[NSLICE][NBLK];     // per-block pos smoothL1 partial sums
  int      tBin[NSLICE];              // threshold bin
  unsigned Rrem[NSLICE];              // elements still needed from threshold bin
  float    negPartial[NSLICE];        // sum of smoothL1 over bins > tBin
  unsigned numNeg[NSLICE];            // min(3*numPos, HW-1)
  unsigned blkCnt[NSLICE][NBLK];      // per-block count of threshold-bin members
  float    inbinPart[NSLICE][NBLK];   // per-block selected in-bin smoothL1 sums
};

// ---------------- helpers ----------------
__device__ __forceinline__ float smoothL1(float a) {  // a = |d| >= 0
  return (a < 1.f) ? 0.5f * a * a : a - 0.5f;
}

// Ranking key = pos ? 0 : |yt-yp|; keys >= 0 so raw float bits are monotone.
__device__ __forceinline__ int keyBin(float tv, float pv, float& sl, bool& pos) {
  float a = fabsf(tv - pv);
  sl  = smoothL1(a);
  pos = (tv >= 0.5f);
  unsigned k = pos ? 0u : __float_as_uint(a);
  return (int)(k >> 19);   // < 2048
}

// ---------------- K1: streaming histogram + positive stats ----------------
__global__ __launch_bounds__(TPB)
void k_hist(const float* __restrict__ yt, const float* __restrict__ yp,
            WS* __restrict__ ws) {
  __shared__ unsigned shC[NBINS];
  __shared__ float    shS[NBINS];
  __shared__ unsigned wN[TPB / 32];
  __shared__ float    wP[TPB / 32];

  const int s   = blockIdx.x / NBLK;
  const int blk = blockIdx.x % NBLK;
  const int tid = threadIdx.x;

  for (int i = tid; i < NBINS; i += TPB) { shC[i] = 0u; shS[i] = 0.f; }
  __syncthreads();

  const size_t base = (size_t)s * HW + (size_t)blk * CHUNK;
  unsigned npos = 0, nbin0 = 0;
  float    psum = 0.f;

  const int ITERS = CHUNK / (TPB * 4);
  for (int it = 0; it < ITERS; ++it) {
    const size_t off = base + (size_t)it * (TPB * 4) + (size_t)tid * 4;
    if (it + 2 < ITERS) {                                  // stay inside the buffer:
      __builtin_prefetch(yt + off + 2 * (TPB * 4), 0, 1);  // prefetch is non-speculative
      __builtin_prefetch(yp + off + 2 * (TPB * 4), 0, 1);  // at WGP scope on CDNA5
    }
    const float4 t4 = *(const float4*)(yt + off);          // global_load_b128
    const float4 p4 = *(const float4*)(yp + off);
    const float tv[4] = {t4.x, t4.y, t4.z, t4.w};
    const float pv[4] = {p4.x, p4.y, p4.z, p4.w};
#pragma unroll
    for (int j = 0; j < 4; ++j) {
      float sl; bool pos;
      const int bn = keyBin(tv[j], pv[j], sl, pos);
      if (pos)          { ++npos; psum += sl; ++nbin0; }  // key==0 -> bin 0, no atomic
      else if (bn == 0) { ++nbin0; }                      // tiny losses: bin 0, no atomic
      else { atomicAdd(&shC[bn], 1u); atomicAdd(&shS[bn], sl); } // ds_add_u32/f32
    }
  }
  if (nbin0) atomicAdd(&shC[0], nbin0);

  // wave32 reduce of positive stats
  for (int o = 16; o; o >>= 1) {
    npos += __shfl_down(npos, o, 32);
    psum += __shfl_down(psum, o, 32);
  }
  const int wid = tid >> 5, lane = tid & 31;
  if (lane == 0) { wN[wid] = npos; wP[wid] = psum; }
  __syncthreads();

  // merge LDS histogram into per-slice global histogram
  for (int i = tid; i < NBINS; i += TPB) {
    const unsigned c = shC[i];
    if (c) atomicAdd(&ws->histC[s][i], c);
    const float v = shS[i];
    if (v != 0.f) atomicAdd(&ws->histS[s][i], v);
  }
  if (tid == 0) {
    unsigned n = 0; float pp = 0.f;
    for (int i = 0; i < TPB / 32; ++i) { n += wN[i]; pp += wP[i]; }
    atomicAdd(&ws->numPos[s], n);       // integer atomics: deterministic
    ws->posPart[s][blk] = pp;           // per-block partial: deterministic final sum
  }
}

// ---------------- K2: parallel per-slice threshold (suffix scan) ----------------
// One 256-thread block per slice. Finds the unique bin b (descending key order)
// with  G(b) = #elements in bins > b  satisfying  G(b) <= num_neg < G(b)+c(b),
// via an inclusive SUFFIX scan S(b); also reduces negPartial deterministically.
__global__ __launch_bounds__(TPB)
void k_threshold(WS* __restrict__ ws) {
  __shared__ unsigned sCnt[NBINS];
  __shared__ unsigned segSuf[TPB];
  __shared__ int      sTb;
  __shared__ unsigned sR;
  __shared__ float    wF[TPB / 32];

  const int s = blockIdx.x, tid = threadIdx.x;

  for (int i = tid; i < NBINS; i += TPB) sCnt[i] = ws->histC[s][i];

  const unsigned np = ws->numPos[s];
  unsigned nneg = (unsigned)NEG_POS * np;
  if (nneg > (unsigned)(HW - 1)) nneg = (unsigned)(HW - 1);
  __syncthreads();

  // local suffix sums inside this thread's 8-bin segment [tid*8, tid*8+8)
  unsigned loc[BPT];
  unsigned segTot = 0;
  for (int j = BPT - 1; j >= 0; --j) { segTot += sCnt[tid * BPT + j]; loc[j] = segTot; }
  segSuf[tid] = segTot;
  __syncthreads();

  // Hillis-Steele inclusive suffix scan over the 256 segment totals (8 steps)
  for (int o = 1; o < TPB; o <<= 1) {
    const unsigned v = (tid + o < TPB) ? segSuf[tid + o] : 0u;
    __syncthreads();
    segSuf[tid] += v;
    __syncthreads();
  }
  const unsigned after = (tid + 1 < TPB) ? segSuf[tid + 1] : 0u;

  // locate the crossing bin: S(b) > nneg >= S(b) - c(b)  (unique; requires c>0)
  for (int j = 0; j < BPT; ++j) {
    const unsigned Sb = loc[j] + after;            // inclusive suffix at bin b
    const unsigned c  = sCnt[tid * BPT + j];
    if (Sb > nneg && (Sb - c) <= nneg) {
      sTb = tid * BPT + j;
      sR  = nneg - (Sb - c);
    }
  }
  __syncthreads();
  const int tb = sTb;

  // negPartial = sum of smoothL1 over fully-selected bins (> tb); fixed-order reduce
  float acc = 0.f;
  for (int i = tid; i < NBINS; i += TPB)
    if (i > tb) acc += ws->histS[s][i];
  for (int o = 16; o; o >>= 1) acc += __shfl_down(acc, o, 32);
  const int wid = tid >> 5, lane = tid & 31;
  if (lane == 0) wF[wid] = acc;
  __syncthreads();
  if (tid == 0) {
    float tot = 0.f;
    for (int i = 0; i < TPB / 32; ++i) tot += wF[i];
    ws->negPartial[s] = tot;
    ws->numNeg[s]     = nneg;
    ws->tBin[s]       = tb;
    ws->Rrem[s]       = sR;
  }
}

// ---------------- K3: per-block count of threshold-bin members (L2-resident) ----
__global__ __launch_bounds__(TPB)
void k_count(const float* __restrict__ yt, const float* __restrict__ yp,
             WS* __restrict__ ws) {
  __shared__ unsigned wC[TPB / 32];
  const int s   = blockIdx.x / NBLK;
  const int blk = blockIdx.x % NBLK;
  const int tid = threadIdx.x;
  const int tb  = ws->tBin[s];
  const size_t base = (size_t)s * HW + (size_t)blk * CHUNK;

  unsigned cnt = 0;
  for (int it = 0; it < CHUNK / (TPB * 4); ++it) {
    const size_t off = base + (size_t)it * (TPB * 4) + (size_t)tid * 4;
    const float4 t4 = *(const float4*)(yt + off);
    const float4 p4 = *(const float4*)(yp + off);
    const float tv[4] = {t4.x, t4.y, t4.z, t4.w};
    const float pv[4] = {p4.x, p4.y, p4.z, p4.w};
#pragma unroll
    for (int j = 0; j < 4; ++j) {
      float sl; bool pos;
      cnt += (keyBin(tv[j], pv[j], sl, pos) == tb) ? 1u : 0u;
    }
  }
  for (int o = 16; o; o >>= 1) cnt += __shfl_down(cnt, o, 32);
  const int wid = tid >> 5, lane = tid & 31;
  if (lane == 0) wC[wid] = cnt;
  __syncthreads();
  if (tid == 0) {
    unsigned tot = 0;
    for (int i = 0; i < TPB / 32; ++i) tot += wC[i];
    ws->blkCnt[s][blk] = tot;
  }
}

// ---------------- K5: select first R in-bin members in index order ------------
__global__ __launch_bounds__(TPB)
void k_select(const float* __restrict__ yt, const float* __restrict__ yp,
              WS* __restrict__ ws) {
  __shared__ float wF[TPB / 32];
  const int s   = blockIdx.x / NBLK;
  const int blk = blockIdx.x % NBLK;
  const int tid = threadIdx.x;
  const int tb  = ws->tBin[s];
  const unsigned R = ws->Rrem[s];

  unsigned S0 = 0;
  for (int b = 0; b < blk; ++b) S0 += ws->blkCnt[s][b];
  const unsigned myCnt = ws->blkCnt[s][blk];
  if (S0 >= R) return;                       // block entirely past the cut

  const size_t base = (size_t)s * HW + (size_t)blk * CHUNK;

  if (S0 + myCnt <= R) {                     // block entirely selected: any order
    float acc = 0.f;
    for (int it = 0; it < CHUNK / (TPB * 4); ++it) {
      const size_t off = base + (size_t)it * (TPB * 4) + (size_t)tid * 4;
      const float4 t4 = *(const float4*)(yt + off);
      const float4 p4 = *(const float4*)(yp + off);
      const float tv[4] = {t4.x, t4.y, t4.z, t4.w};
      const float pv[4] = {p4.x, p4.y, p4.z, p4.w};
#pragma unroll
      for (int j = 0; j < 4; ++j) {
        float sl; bool pos;
        if (keyBin(tv[j], pv[j], sl, pos) == tb) acc += sl;
      }
    }
    for (int o = 16; o; o >>= 1) acc += __shfl_down(acc, o, 32);
    const int wid = tid >> 5, lane = tid & 31;
    if (lane == 0) wF[wid] = acc;
    __syncthreads();
    if (tid == 0) {
      float tot = 0.f;
      for (int i = 0; i < TPB / 32; ++i) tot += wF[i];
      ws->inbinPart[s][blk] = tot;
    }
    return;
  }

  // Straddling block (one per slice): ordered wave32 scan in ascending index.
  if (tid >= 32) return;
  const unsigned need = R - S0;              // 1 <= need < myCnt
  const int lane = tid;
  unsigned got = 0; float acc = 0.f;
  for (int it = 0; it < CHUNK / 128 && got < need; ++it) {
    const size_t off = base + (size_t)it * 128 + (size_t)lane * 4;
    const float4 t4 = *(const float4*)(yt + off);
    const float4 p4 = *(const float4*)(yp + off);
    const float tv[4] = {t4.x, t4.y, t4.z, t4.w};
    const float pv[4] = {p4.x, p4.y, p4.z, p4.w};
    unsigned m[4]; float slv[4]; unsigned c = 0;
#pragma unroll
    for (int j = 0; j < 4; ++j) {
      float sl; bool pos;
      const bool hit = (keyBin(tv[j], pv[j], sl, pos) == tb);
      m[j] = hit ? 1u : 0u; slv[j] = sl; c += m[j];
    }
    unsigned inc = c;                         // inclusive scan over lanes
    for (int o = 1; o < 32; o <<= 1) {
      const unsigned v = __shfl_up(inc, o, 32);
      if (lane >= o) inc += v;
    }
    unsigned rank = got + (inc - c);          // index-order rank of my first hit
#pragma unroll
    for (int j = 0; j < 4; ++j) {
      if (m[j]) { if (rank < need) acc += slv[j]; ++rank; }
    }
    got += __shfl(inc, 31, 32);
  }
  for (int o = 16; o; o >>= 1) acc += __shfl_down(acc, o, 32);
  if (lane == 0) ws->inbinPart[s][blk] = acc;
}

// ---------------- K6: final combine; wave sum via V_WMMA_F32_16X16X4_F32 ------
__device__ __forceinline__ float wmma_wave_sum(float v) {
#if __has_builtin(__builtin_amdgcn_wmma_f32_16x16x4_f32)
  // A (16x4): lane m holds A[m][0]=v (lanes 0-15) / A[m][2]=v (lanes 16-31); rest 0.
  // B (4x16) = ones  =>  D[m][n] = row-sum of A. Total = sum of 16 row sums.
  v2f a; a.x = v;   a.y = 0.f;
  v2f b; b.x = 1.f; b.y = 1.f;
  v8f c = {0.f, 0.f, 0.f, 0.f, 0.f, 0.f, 0.f, 0.f};
  c = __builtin_amdgcn_wmma_f32_16x16x4_f32(false, a, false, b, (short)0, c,
                                            false, false);
  const float s8 = c[0] + c[1] + c[2] + c[3] + c[4] + c[5] + c[6] + c[7];
  return __shfl(s8, 0, 32) + __shfl(s8, 16, 32);  // rows 0-7 + rows 8-15
#else
  for (int o = 16; o; o >>= 1) v += __shfl_xor(v, o, 32);
  return v;
#endif
}

__global__ void k_final(const WS* __restrict__ ws, float* __restrict__ out) {
  const int lane = threadIdx.x;               // launched with exactly 32 threads
  float ps = 0.f, ns = 0.f, pc = 0.f, nc = 0.f;
  for (int s = lane; s < NSLICE; s += 32) {   // uniform trip count: EXEC all ones
    float pp = 0.f, ib = 0.f;
    for (int b = 0; b < NBLK; ++b) { pp += ws->posPart[s][b]; ib += ws->inbinPart[s][b]; }
    ps += pp;
    ns += ws->negPartial[s] + ib;
    pc += (float)ws->numPos[s];
    nc += (float)ws->numNeg[s];
  }
  ps = wmma_wave_sum(ps);
  ns = wmma_wave_sum(ns);
  pc = wmma_wave_sum(pc);
  nc = wmma_wave_sum(nc);
  if (lane == 0) out[0] = (float)NEG_POS * (ps / pc) + (ns / nc);
}

// ---------------- launch ----------------
extern "C" void kernel_launch(void* const* d_in, const int* in_sizes, int n_in,
                              void* d_out, int out_size, void* d_ws, size_t ws_size,
                              hipStream_t stream) {
  (void)in_sizes; (void)n_in; (void)out_size; (void)ws_size;
  const float* yt = (const float*)d_in[0];   // y_true
  const float* yp = (const float*)d_in[1];   // y_pred
  WS* ws = (WS*)d_ws;

  hipMemsetAsync(d_ws, 0, sizeof(WS), stream);
  k_hist     <<<dim3(NSLICE * NBLK), dim3(TPB), 0, stream>>>(yt, yp, ws);
  k_threshold<<<dim3(NSLICE),        dim3(TPB), 0, stream>>>(ws);
  k_count    <<<dim3(NSLICE * NBLK), dim3(TPB), 0, stream>>>(yt, yp, ws);
  k_select   <<<dim3(NSLICE * NBLK), dim3(TPB), 0, stream>>>(yt, yp, ws);
  k_final    <<<dim3(1),             dim3(32),  0, stream>>>(ws, (float*)d_out);
}